// GaussianRasterizer_86887188398743
// MI455X (gfx1250) — compile-verified
//
#include <hip/hip_runtime.h>
#include <hip/hip_bf16.h>
#include <stdint.h>

// ---------------------------------------------------------------------------
// Gaussian-splat forward rasterizer for MI455X (gfx1250, wave32).
// Memory-bound segmented scan over a pixel-sorted intersection list:
//   - stream (pixel_id, gs_id) through LDS with CDNA5 async global->LDS
//     copies (ASYNCcnt, s_wait_asynccnt), double-buffered
//   - gather the 7.2MB gaussian table from L2, repacked to float4 to cut
//     scattered request count 3x
//   - one thread per pixel walks its own segment: T *= (1-alpha) is the
//     reference's log-space segmented exclusive cumprod, exactly.
// ---------------------------------------------------------------------------

#define BLK 256          // 8 wave32 per workgroup; block owns 256 consecutive pixels
#define CH  2048         // intersection entries staged per chunk (8KB pix + 8KB gs)

// first index in [lo,hi) with a[idx] >= val   (a sorted ascending)
__device__ __forceinline__ int lower_bound_i32(const int* __restrict__ a,
                                               int lo, int hi, int val) {
    while (lo < hi) {
        int mid = (lo + hi) >> 1;
        if (a[mid] < val) lo = mid + 1; else hi = mid;
    }
    return lo;
}

// CDNA5 async copies: 16B / 4B global -> LDS, tracked by ASYNCcnt.
__device__ __forceinline__ void async_cp16(uint32_t lds_addr, const int* gptr) {
    unsigned long long ga = (unsigned long long)(uintptr_t)gptr;
    asm volatile("global_load_async_to_lds_b128 %0, %1, off"
                 :: "v"(lds_addr), "v"(ga) : "memory");
}
__device__ __forceinline__ void async_cp4(uint32_t lds_addr, const int* gptr) {
    unsigned long long ga = (unsigned long long)(uintptr_t)gptr;
    asm volatile("global_load_async_to_lds_b32 %0, %1, off"
                 :: "v"(lds_addr), "v"(ga) : "memory");
}

// Copy chunk [cs, cs+CH) (clipped to M) of both streams into LDS.
// Full in-bounds chunks: every wave issues exactly 4 async instructions
// (2 iters x 2 arrays of b128) -- the double-buffer wait relies on this.
// The ragged array tail (last chunk only) uses b32 copies.
__device__ __forceinline__ void copy_chunk(int cs, int M,
                                           const int* __restrict__ pix_g,
                                           const int* __restrict__ gs_g,
                                           int* sp, int* sg, int tid) {
#pragma unroll
    for (int it = 0; it < CH / (BLK * 4); ++it) {      // 2 iterations
        int off4 = tid * 4 + it * (BLK * 4);
        int g = cs + off4;
        if (g + 4 <= M) {
            async_cp16((uint32_t)(uintptr_t)&sp[off4], pix_g + g);
            async_cp16((uint32_t)(uintptr_t)&sg[off4], gs_g + g);
        } else {
#pragma unroll
            for (int j = 0; j < 4; ++j) {
                if (g + j < M) {
                    async_cp4((uint32_t)(uintptr_t)&sp[off4 + j], pix_g + g + j);
                    async_cp4((uint32_t)(uintptr_t)&sg[off4 + j], gs_g + g + j);
                }
            }
        }
    }
}

// Repack gaussian attrs into 2x float4 + float: 3 gather loads instead of 9.
__global__ __launch_bounds__(256)
void pack_kernel(const float* __restrict__ means2d, const float* __restrict__ conics,
                 const float* __restrict__ opac, const float* __restrict__ colors,
                 float4* __restrict__ tabA, float4* __restrict__ tabB,
                 float* __restrict__ tabC, int N) {
    int g = blockIdx.x * 256 + threadIdx.x;
    if (g < N) {
        tabA[g] = make_float4(means2d[2 * g], means2d[2 * g + 1],
                              conics[3 * g], conics[3 * g + 1]);
        tabB[g] = make_float4(conics[3 * g + 2], opac[g],
                              colors[3 * g], colors[3 * g + 1]);
        tabC[g] = colors[3 * g + 2];
    }
}

template <bool PACKED>
__global__ __launch_bounds__(BLK)
void gsplat_fwd_kernel(const float4* __restrict__ tabA,      // {mx,my,ca,cb}
                       const float4* __restrict__ tabB,      // {cc,op,cr,cg}
                       const float*  __restrict__ tabC,      // {cb_col}
                       const float*  __restrict__ means2d,   // unpacked fallback
                       const float*  __restrict__ conics,
                       const float*  __restrict__ opac,
                       const float*  __restrict__ colors,
                       const int*    __restrict__ pixel_ids, // [M] sorted
                       const int*    __restrict__ gs_ids,    // [M]
                       const int*    __restrict__ wptr,      // width scalar
                       float*        __restrict__ out,       // [3, H*W]
                       int HW, int M) {
    __shared__ int s_pix[2][CH];
    __shared__ int s_gs[2][CH];
    __shared__ int s_range[2];

    const int tid = threadIdx.x;
    const int p0  = blockIdx.x * BLK;
    const int width = *wptr;

    // Block-uniform slice [s,e) of the sorted intersection list.
    if (tid < 2)
        s_range[tid] = lower_bound_i32(pixel_ids, 0, M, p0 + tid * BLK);
    __syncthreads();
    const int s = s_range[0];
    const int e = s_range[1];
    const int nc = (e - s + CH - 1) / CH;   // uniform per block

    // Kick off the first two chunk copies immediately so they overlap the
    // per-thread binary search below.
    if (nc > 0) copy_chunk(s,      M, pixel_ids, gs_ids, s_pix[0], s_gs[0], tid);
    if (nc > 1) copy_chunk(s + CH, M, pixel_ids, gs_ids, s_pix[1], s_gs[1], tid);

    const int p = p0 + tid;                 // this thread's pixel / ray id
    const float px = (float)(p % width) + 0.5f;
    const float py = (float)(p / width) + 0.5f;

    // This thread's segment starts here (first entry with pixel >= p).
    int i = lower_bound_i32(pixel_ids, s, e, p);

    float T = 1.0f, accr = 0.0f, accg = 0.0f, accb = 0.0f;
    bool done = false;

    for (int k = 0; k < nc; ++k) {
        const int cur = k & 1;
        const int cs  = s + k * CH;
        const int ce  = min(e, cs + CH);

        // Retire chunk k. If the in-flight chunk k+1 is guaranteed full, it
        // contributed exactly 4 async instrs/wave, so leaving 4 outstanding
        // retires chunk k precisely (in-order completion). Otherwise drain.
        if (k + 1 < nc && s + (k + 2) * CH <= e)
            asm volatile("s_wait_asynccnt 0x4" ::: "memory");
        else
            asm volatile("s_wait_asynccnt 0x0" ::: "memory");
        __syncthreads();                    // chunk k visible to all waves

        while (!done && i < ce) {
            const int j = i - cs;
            if (s_pix[cur][j] != p) { done = true; break; }
            const int g = s_gs[cur][j];     // C==1 -> flat gaussian index

            float dx, dy, sig, alpha, cR, cG, cB;
            if (PACKED) {
                const float4 A = tabA[g];
                const float4 B = tabB[g];
                cB = tabC[g];
                dx = px - A.x; dy = py - A.y;
                sig = 0.5f * (A.z * dx * dx + B.x * dy * dy) + A.w * dx * dy;
                alpha = fminf(B.y * expf(-sig), 0.999f);
                cR = B.z; cG = B.w;
            } else {
                dx = px - means2d[2 * g]; dy = py - means2d[2 * g + 1];
                const float ca = conics[3 * g];
                const float cb = conics[3 * g + 1];
                const float cc = conics[3 * g + 2];
                sig = 0.5f * (ca * dx * dx + cc * dy * dy) + cb * dx * dy;
                alpha = fminf(opac[g] * expf(-sig), 0.999f);
                cR = colors[3 * g]; cG = colors[3 * g + 1]; cB = colors[3 * g + 2];
            }

            const float w = alpha * T;      // alpha * exclusive prod(1-alpha)
            accr += w * cR;
            accg += w * cG;
            accb += w * cB;
            T *= (1.0f - alpha);
            ++i;
        }
        __syncthreads();                    // all readers of buf[cur] done

        // Refill the buffer chunk k occupied with chunk k+2 (safe: barrier
        // above guarantees no wave still reads it; its own copies retired
        // before processing began).
        if (k + 2 < nc)
            copy_chunk(s + (k + 2) * CH, M, pixel_ids, gs_ids,
                       s_pix[cur], s_gs[cur], tid);
    }

    if (p < HW) {                           // [3,H,W] channel-major output
        out[p]          = accr;
        out[HW + p]     = accg;
        out[2 * HW + p] = accb;
    }
}

extern "C" void kernel_launch(void* const* d_in, const int* in_sizes, int n_in,
                              void* d_out, int out_size, void* d_ws, size_t ws_size,
                              hipStream_t stream) {
    (void)n_in;
    const float* means2d   = (const float*)d_in[0];
    const float* conics    = (const float*)d_in[1];
    const float* opac      = (const float*)d_in[2];
    const float* colors    = (const float*)d_in[3];
    const int*   pixel_ids = (const int*)d_in[4];
    const int*   gs_ids    = (const int*)d_in[5];
    // d_in[6] camera_ids: all zero (C==1) -> unused
    // d_in[7] height: not needed (py derives from pixel_id / width)
    const int*   wptr      = (const int*)d_in[8];
    float*       out       = (float*)d_out;

    const int M     = in_sizes[4];
    const int Npack = in_sizes[2];          // C*N gaussians
    const int HW    = out_size / 3;
    const int grid  = (HW + BLK - 1) / BLK;

    const size_t needed = (size_t)Npack * 36;   // float4 + float4 + float
    if (ws_size >= needed) {
        float4* tabA = (float4*)d_ws;
        float4* tabB = (float4*)((char*)d_ws + (size_t)Npack * 16);
        float*  tabC = (float*) ((char*)d_ws + (size_t)Npack * 32);
        pack_kernel<<<(Npack + 255) / 256, 256, 0, stream>>>(
            means2d, conics, opac, colors, tabA, tabB, tabC, Npack);
        gsplat_fwd_kernel<true><<<grid, BLK, 0, stream>>>(
            tabA, tabB, tabC, means2d, conics, opac, colors,
            pixel_ids, gs_ids, wptr, out, HW, M);
    } else {
        gsplat_fwd_kernel<false><<<grid, BLK, 0, stream>>>(
            nullptr, nullptr, nullptr, means2d, conics, opac, colors,
            pixel_ids, gs_ids, wptr, out, HW, M);
    }
}